// SRLEmbeddings_89464168776083
// MI455X (gfx1250) — compile-verified
//
#include <hip/hip_runtime.h>

typedef __attribute__((ext_vector_type(2))) float v2f;
typedef __attribute__((ext_vector_type(8))) float v8f;

#define BATCH 64
#define SEQ   512
#define DIM   768
#define NT    16
#define NL    10
#define WROWS 48
#define WSTRIDE 516   // 516 % 64 == 4 -> dual-half A fetch covers all 64 banks, conflict-free
#define PF    24      // prefetch distance (rows of emb); speculative -> safe past the end

// Fused SRL-embedding kernel:
//   out48[r, d] = sum_s W[r, s] * emb[b, s, d]   (rows: 0-15 pred, 16-31 arg0, 32-47 arg1)
//   sentence average accumulated on the side from the streamed B operands.
// Exact f32 arithmetic via V_WMMA_F32_16X16X4_F32.
__global__ __launch_bounds__(128) void srl_wmma_kernel(
    const float* __restrict__ emb,       // [B, S, D]
    const int*   __restrict__ pred_ids,  // [B, T, L]
    const int*   __restrict__ arg0_ids,  // [B, T, L]
    const int*   __restrict__ arg1_ids,  // [B, T, L]
    float*       __restrict__ out)       // avg [B,D] ++ pred [B,T,D] ++ a0 ++ a1
{
    extern __shared__ float W[];         // [48][WSTRIDE] ~ 97 KB -> 3 workgroups / WGP
    const int b   = blockIdx.x;
    const int tid = threadIdx.x;

    // ---- build per-batch scatter-weight matrix W in LDS ----
    for (int i = tid; i < WROWS * WSTRIDE; i += 128) W[i] = 0.0f;
    __syncthreads();
    if (tid < 48) {                      // one thread per row: race-free accumulation
        const int q = tid >> 4, t = tid & 15;
        const int* ids = (q == 0 ? pred_ids : (q == 1 ? arg0_ids : arg1_ids))
                         + ((size_t)b * NT + t) * NL;
        int cnt = 0;
        for (int l = 0; l < NL; ++l) {
            int id = ids[l];
            if (id != 0 && id < SEQ) ++cnt;
        }
        if (cnt > 0) {
            const float w = 1.0f / (float)cnt;
            float* row = &W[tid * WSTRIDE];
            for (int l = 0; l < NL; ++l) {
                int id = ids[l];
                if (id != 0 && id < SEQ) row[id] += w;   // duplicates accumulate, as in ref
            }
        }
    }
    __syncthreads();

    // ---- WMMA main loop: each wave owns TWO adjacent 16-col N-tiles ----
    const int wave = tid >> 5;
    const int lane = tid & 31;
    const int n0   = (blockIdx.y * 8 + wave * 2) * 16;  // 6 blk * 4 waves * 2 tiles * 16 = 768
    const int n    = lane & 15;                         // A row m / B,D column N
    const int h    = lane >> 4;                         // lane half: K offset = 2h

    v8f acc00 = {}, acc01 = {};   // pred  x {tile0, tile1}
    v8f acc10 = {}, acc11 = {};   // arg0
    v8f acc20 = {}, acc21 = {};   // arg1
    float sax = 0.0f, say = 0.0f; // sentence-sum partials for tile0 / tile1

    const float* e0 = emb + (size_t)b * SEQ * DIM + n0 + n;
    const float* e1 = e0 + 16;

    for (int s0 = 0; s0 < SEQ; s0 += 4) {
        const int k = s0 + 2 * h;
        // B 4x16 f32 tiles: vgpr j holds emb[b, k + j, col]
        const float* p0 = e0 + (size_t)k * DIM;
        const float* p1 = e1 + (size_t)k * DIM;
        v2f b0; b0.x = p0[0]; b0.y = p0[DIM];
        v2f b1; b1.x = p1[0]; b1.y = p1[DIM];
        // unconditional speculative prefetch of the full 128B lines, PF rows ahead
        // (covers rows k+PF .. k+PF+3 across the two lane halves; dropped silently if OOB)
        __builtin_prefetch(p0 + (size_t)PF * DIM, 0, 3);
        __builtin_prefetch(p0 + (size_t)(PF + 1) * DIM, 0, 3);
        sax += b0.x + b0.y;             // sentence average rides along for free
        say += b1.x + b1.y;
        // A 16x4 f32 tiles from LDS (column-adjacent pair -> ds_load_2addr/b64)
        const float* w0 = &W[( 0 + n) * WSTRIDE + k];
        const float* w1 = &W[(16 + n) * WSTRIDE + k];
        const float* w2 = &W[(32 + n) * WSTRIDE + k];
        v2f a0; a0.x = w0[0]; a0.y = w0[1];
        v2f a1; a1.x = w1[0]; a1.y = w1[1];
        v2f a2; a2.x = w2[0]; a2.y = w2[1];
        acc00 = __builtin_amdgcn_wmma_f32_16x16x4_f32(false, a0, false, b0, (short)0, acc00, false, false);
        acc01 = __builtin_amdgcn_wmma_f32_16x16x4_f32(false, a0, false, b1, (short)0, acc01, false, false);
        acc10 = __builtin_amdgcn_wmma_f32_16x16x4_f32(false, a1, false, b0, (short)0, acc10, false, false);
        acc11 = __builtin_amdgcn_wmma_f32_16x16x4_f32(false, a1, false, b1, (short)0, acc11, false, false);
        acc20 = __builtin_amdgcn_wmma_f32_16x16x4_f32(false, a2, false, b0, (short)0, acc20, false, false);
        acc21 = __builtin_amdgcn_wmma_f32_16x16x4_f32(false, a2, false, b1, (short)0, acc21, false, false);
    }

    // ---- store: C/D layout vgpr v -> row M = v + 8h, col N ----
    float* out_avg = out;                               // [B, D]
    float* out_ex  = out + (size_t)BATCH * DIM;         // 3 x [B, T, D]
    const size_t extent = (size_t)BATCH * NT * DIM;
    const int M0 = 8 * h;
    #pragma unroll
    for (int v = 0; v < 8; ++v) {
        const int t = v + M0;
        const size_t o0 = ((size_t)(b * NT + t)) * DIM + n0 + n;
        const size_t o1 = o0 + 16;
        out_ex[0 * extent + o0] = acc00[v];
        out_ex[0 * extent + o1] = acc01[v];
        out_ex[1 * extent + o0] = acc10[v];
        out_ex[1 * extent + o1] = acc11[v];
        out_ex[2 * extent + o0] = acc20[v];
        out_ex[2 * extent + o1] = acc21[v];
    }
    // sentence average: combine the two lane halves (h=0 holds s%4 in {0,1}, h=1 in {2,3})
    const float ox = __shfl_xor(sax, 16, 32);
    const float oy = __shfl_xor(say, 16, 32);
    if (h == 0) {
        out_avg[(size_t)b * DIM + n0 + n]      = (sax + ox) * (1.0f / (float)SEQ);
        out_avg[(size_t)b * DIM + n0 + 16 + n] = (say + oy) * (1.0f / (float)SEQ);
    }
}

extern "C" void kernel_launch(void* const* d_in, const int* in_sizes, int n_in,
                              void* d_out, int out_size, void* d_ws, size_t ws_size,
                              hipStream_t stream) {
    const float* emb  = (const float*)d_in[0];
    const int*   pred = (const int*)d_in[1];
    const int*   a0   = (const int*)d_in[2];
    const int*   a1   = (const int*)d_in[3];
    float*       out  = (float*)d_out;
    (void)in_sizes; (void)n_in; (void)out_size; (void)d_ws; (void)ws_size;

    dim3 grid(BATCH, 6), block(128);
    size_t shmem = (size_t)WROWS * WSTRIDE * sizeof(float);   // ~97 KB
    hipLaunchKernelGGL(srl_wmma_kernel, grid, block, shmem, stream,
                       emb, pred, a0, a1, out);
}